// Concat_26139170963895
// MI455X (gfx1250) — compile-verified
//
#include <hip/hip_runtime.h>

#define BATCH 16
#define CHN   3
#define IMH   512
#define IMW   512
#define HW    (IMH * IMW)
#define KSEL  26          // int(1e-4 * 512 * 512)
#define HALO  7           // (15-1)/2
#define TILE  32
#define IN_T  (TILE + 2 * HALO)   // 46
#define LDS_STRIDE 48
#define NTH   256
#define PERT  (HW / NTH)          // 1024
#define NLOAD 9                   // ceil(IN_T*IN_T / NTH) = ceil(2116/256)

// ---- async global -> LDS staging of a clamped (clamp-to-edge) halo tile ----
// Clamp-to-edge is exactly equivalent to SAME/-inf padding for max AND min
// pooling because the clamped coordinate always lies inside the window.
// Every thread issues exactly NLOAD async loads (tail indices clamped to the
// last tile element -> duplicate same-value LDS writes, harmless), so each
// wave's ASYNCcnt contribution per tile is a compile-time constant and we can
// software-pipeline with  s_wait_asynccnt <= NLOAD.
__device__ __forceinline__ void issue_tile_loads(const float* __restrict__ src,
                                                 float (*tile)[LDS_STRIDE],
                                                 int ty0, int tx0, int tid) {
#pragma unroll
    for (int k = 0; k < NLOAD; ++k) {
        int i = tid + k * NTH;
        if (i >= IN_T * IN_T) i = IN_T * IN_T - 1;
        int ly = i / IN_T, lx = i - ly * IN_T;
        int gy = min(max(ty0 - HALO + ly, 0), IMH - 1);
        int gx = min(max(tx0 - HALO + lx, 0), IMW - 1);
        unsigned long long gp = (unsigned long long)(src + (size_t)gy * IMW + gx);
        unsigned lds_addr = (unsigned)(unsigned long long)(&tile[ly][lx]);
        asm volatile("global_load_async_to_lds_b32 %0, %1, off"
                     :: "v"(lds_addr), "v"(gp) : "memory");
    }
}

__device__ __forceinline__ void wait_async_pending(void) {
    asm volatile("s_wait_asynccnt %0" :: "i"(NLOAD) : "memory");
}
__device__ __forceinline__ void wait_async_all(void) {
    asm volatile("s_wait_asynccnt 0x0" ::: "memory");
}

// ---- Stage 1: dark/bright channel maps via separable 15-tap pooling --------
// Double-buffered channel pipeline: prefetch channel c+1 while pooling c.
__global__ __launch_bounds__(NTH)
void pool_dcbc_kernel(const float* __restrict__ x,
                      float* __restrict__ dc, float* __restrict__ bc) {
    __shared__ float tile[2][IN_T][LDS_STRIDE];
    __shared__ float hmax[IN_T][TILE];
    __shared__ float hmin[IN_T][TILE];
    const int b   = blockIdx.z;
    const int ty0 = blockIdx.y * TILE;
    const int tx0 = blockIdx.x * TILE;
    const int tid = threadIdx.x;
    const float* xb = x + (size_t)b * CHN * HW;

    float dcv[4], bcv[4];
#pragma unroll
    for (int r = 0; r < 4; ++r) { dcv[r] = 1e30f; bcv[r] = -1e30f; }

    issue_tile_loads(xb, tile[0], ty0, tx0, tid);   // prefetch channel 0

    for (int c = 0; c < CHN; ++c) {
        if (c + 1 < CHN) {
            issue_tile_loads(xb + (size_t)(c + 1) * HW, tile[(c + 1) & 1],
                             ty0, tx0, tid);        // prefetch next channel
            wait_async_pending();                   // current tile done
        } else {
            wait_async_all();
        }
        __syncthreads();                            // tile[c&1] ready everywhere
        const float (*cur)[LDS_STRIDE] = tile[c & 1];

        // horizontal 15-tap max & min: IN_T rows x TILE cols
        for (int i = tid; i < IN_T * TILE; i += NTH) {
            int ly = i / TILE, lx = i - ly * TILE;
            float mx = cur[ly][lx], mn = mx;
#pragma unroll
            for (int d = 1; d < 15; ++d) {
                float v = cur[ly][lx + d];
                mx = fmaxf(mx, v); mn = fminf(mn, v);
            }
            hmax[ly][lx] = mx; hmin[ly][lx] = mn;
        }
        __syncthreads();

        // vertical 15-tap, 4 outputs per thread
#pragma unroll
        for (int r = 0; r < 4; ++r) {
            int o = tid + r * NTH;
            int oy = o / TILE, ox = o - oy * TILE;
            float mx = hmax[oy][ox], mn = hmin[oy][ox];
#pragma unroll
            for (int d = 1; d < 15; ++d) {
                mx = fmaxf(mx, hmax[oy + d][ox]);
                mn = fminf(mn, hmin[oy + d][ox]);
            }
            bcv[r] = fmaxf(bcv[r], mx);
            dcv[r] = fminf(dcv[r], mn);
        }
        __syncthreads();
    }

#pragma unroll
    for (int r = 0; r < 4; ++r) {
        int o = tid + r * NTH;
        int oy = o / TILE, ox = o - oy * TILE;
        size_t off = (size_t)b * HW + (size_t)(ty0 + oy) * IMW + (tx0 + ox);
        dc[off] = dcv[r];
        bc[off] = bcv[r];
    }
}

// ---- Stage 2: exact top-26 per batch, fully LDS-resident -------------------
// mode 0: top-k of dc -> compute A1[b][3] directly
// mode 1: top-k of -bc -> write 26 indices to idxout[b*26..]
__global__ __launch_bounds__(NTH)
void topk_kernel(const float* __restrict__ vals, int negate,
                 const float* __restrict__ x,
                 float* __restrict__ A1, int* __restrict__ idxout) {
    __shared__ float lv[NTH][KSEL];
    __shared__ int   li[NTH][KSEL];
    __shared__ float rv[NTH];
    __shared__ int   ri[NTH];
    __shared__ int   sel[KSEL];

    const int b = blockIdx.x, tid = threadIdx.x;
    const float* vb = vals + (size_t)b * HW;
    const float sgn = negate ? -1.0f : 1.0f;

    // seed local list with first KSEL slice elements
    for (int k = 0; k < KSEL; ++k) {
        int i = tid + k * NTH;
        lv[tid][k] = sgn * vb[i];
        li[tid][k] = i;
    }
    float mnv = lv[tid][0]; int mns = 0;
    for (int k = 1; k < KSEL; ++k)
        if (lv[tid][k] < mnv) { mnv = lv[tid][k]; mns = k; }

    // replace-min streaming top-26 over remaining slice elements
    for (int j = KSEL; j < PERT; ++j) {
        int i = tid + j * NTH;
        float v = sgn * vb[i];
        if (v > mnv) {
            lv[tid][mns] = v; li[tid][mns] = i;
            mnv = lv[tid][0]; mns = 0;
            for (int k = 1; k < KSEL; ++k)
                if (lv[tid][k] < mnv) { mnv = lv[tid][k]; mns = k; }
        }
    }
    __syncthreads();

    // 26 rounds of extract-max across the 256 local lists
    for (int r = 0; r < KSEL; ++r) {
        float m = lv[tid][0]; int ms = 0;
        for (int k = 1; k < KSEL; ++k)
            if (lv[tid][k] > m) { m = lv[tid][k]; ms = k; }
        rv[tid] = m; ri[tid] = (tid << 5) | ms;
        __syncthreads();
        for (int s = NTH / 2; s > 0; s >>= 1) {
            if (tid < s && rv[tid + s] > rv[tid]) {
                rv[tid] = rv[tid + s]; ri[tid] = ri[tid + s];
            }
            __syncthreads();
        }
        if (tid == 0) {
            int code = ri[0], owner = code >> 5, slot = code & 31;
            sel[r] = li[owner][slot];
            lv[owner][slot] = -1e30f;
        }
        __syncthreads();
    }

    if (!negate) {
        if (tid == 0) {
            const float* xb = x + (size_t)b * CHN * HW;
            float bestpm = -1e30f; int bi = sel[0];
            for (int r = 0; r < KSEL; ++r) {
                int i = sel[r];
                float pm = fmaxf(xb[i], fmaxf(xb[HW + i], xb[2 * HW + i]));
                if (pm > bestpm) { bestpm = pm; bi = i; }
            }
            A1[b * 3 + 0] = xb[bi];
            A1[b * 3 + 1] = xb[HW + bi];
            A1[b * 3 + 2] = xb[2 * HW + bi];
        }
    } else {
        if (tid < KSEL) idxout[b * KSEL + tid] = sel[tid];
    }
}

// ---- Stage 3: A2 (over ALL batches' indices — reference bug), A, s ---------
__global__ void ambient_kernel(const float* __restrict__ x,
                               const float* __restrict__ A1,
                               const int* __restrict__ idx2,
                               float* __restrict__ s) {
    int t = threadIdx.x;
    if (t >= BATCH * CHN) return;
    int b = t / CHN, c = t - b * CHN;
    const float* xb = x + ((size_t)b * CHN + c) * HW;
    float acc = 0.0f;
    for (int j = 0; j < BATCH * KSEL; ++j) acc += xb[idx2[j]];
    float A2 = acc / (float)(BATCH * KSEL);
    float A = 0.75f * A1[t] + 0.25f * A2;
    s[t] = 1.0f / (1.0f - A + 1e-6f);
}

// ---- Stage 4: t = 1 - 0.95 * min_c s_c*(1 - maxpool15(x_c)); concat --------
__global__ __launch_bounds__(NTH)
void final_kernel(const float* __restrict__ x, const float* __restrict__ sA,
                  float* __restrict__ out) {
    __shared__ float tile[2][IN_T][LDS_STRIDE];
    __shared__ float hmax[IN_T][TILE];
    const int b   = blockIdx.z;
    const int ty0 = blockIdx.y * TILE;
    const int tx0 = blockIdx.x * TILE;
    const int tid = threadIdx.x;
    const float* xb = x + (size_t)b * CHN * HW;

    float tmin[4];
#pragma unroll
    for (int r = 0; r < 4; ++r) tmin[r] = 1e30f;

    issue_tile_loads(xb, tile[0], ty0, tx0, tid);   // prefetch channel 0

    for (int c = 0; c < CHN; ++c) {
        const float sc = sA[b * 3 + c];
        if (c + 1 < CHN) {
            issue_tile_loads(xb + (size_t)(c + 1) * HW, tile[(c + 1) & 1],
                             ty0, tx0, tid);
            wait_async_pending();
        } else {
            wait_async_all();
        }
        __syncthreads();
        const float (*cur)[LDS_STRIDE] = tile[c & 1];

        // copy x into out channel c straight from LDS (center never clamped)
#pragma unroll
        for (int r = 0; r < 4; ++r) {
            int o = tid + r * NTH;
            int oy = o / TILE, ox = o - oy * TILE;
            out[(((size_t)b * 4 + c) * IMH + ty0 + oy) * IMW + tx0 + ox] =
                cur[HALO + oy][HALO + ox];
        }

        for (int i = tid; i < IN_T * TILE; i += NTH) {
            int ly = i / TILE, lx = i - ly * TILE;
            float mx = cur[ly][lx];
#pragma unroll
            for (int d = 1; d < 15; ++d) mx = fmaxf(mx, cur[ly][lx + d]);
            hmax[ly][lx] = mx;
        }
        __syncthreads();

#pragma unroll
        for (int r = 0; r < 4; ++r) {
            int o = tid + r * NTH;
            int oy = o / TILE, ox = o - oy * TILE;
            float mx = hmax[oy][ox];
#pragma unroll
            for (int d = 1; d < 15; ++d) mx = fmaxf(mx, hmax[oy + d][ox]);
            tmin[r] = fminf(tmin[r], (1.0f - mx) * sc);
        }
        __syncthreads();
    }

#pragma unroll
    for (int r = 0; r < 4; ++r) {
        int o = tid + r * NTH;
        int oy = o / TILE, ox = o - oy * TILE;
        out[(((size_t)b * 4 + 3) * IMH + ty0 + oy) * IMW + tx0 + ox] =
            1.0f - 0.95f * tmin[r];
    }
}

extern "C" void kernel_launch(void* const* d_in, const int* in_sizes, int n_in,
                              void* d_out, int out_size, void* d_ws, size_t ws_size,
                              hipStream_t stream) {
    (void)in_sizes; (void)n_in; (void)out_size; (void)ws_size;
    const float* x = (const float*)d_in[0];
    float* out = (float*)d_out;

    float* dc   = (float*)d_ws;                 // 16 MB
    float* bc   = dc + (size_t)BATCH * HW;      // 16 MB
    float* A1   = bc + (size_t)BATCH * HW;      // 48 floats
    float* sA   = A1 + BATCH * CHN;             // 48 floats
    int*   idx2 = (int*)(sA + BATCH * CHN);     // 416 ints

    dim3 grid(IMW / TILE, IMH / TILE, BATCH);
    pool_dcbc_kernel<<<grid, NTH, 0, stream>>>(x, dc, bc);
    topk_kernel<<<BATCH, NTH, 0, stream>>>(dc, 0, x, A1, idx2);
    topk_kernel<<<BATCH, NTH, 0, stream>>>(bc, 1, x, A1, idx2);
    ambient_kernel<<<1, 64, 0, stream>>>(x, A1, idx2, sA);
    final_kernel<<<grid, NTH, 0, stream>>>(x, sA, out);
}